// ParallelShortHyenaOperator_38774964748500
// MI455X (gfx1250) — compile-verified
//
#include <hip/hip_runtime.h>
#include <cstdint>

namespace {
constexpr int kB   = 2;
constexpr int kL   = 4096;
constexpr int kG   = 256;
constexpr int kDG  = 16;
constexpr int kD   = kG * kDG;       // 4096 channels
constexpr int kK   = 7;              // conv taps
constexpr int kTilesPerSeg = 32;     // 32 tiles x 16 rows = 512 L rows per wave
constexpr int kSegRows = kTilesPerSeg * 16;
constexpr int kSegs = kL / kSegRows; // 8
constexpr int kWavesPerBlock = 8;
constexpr int kTPB = kWavesPerBlock * 32;
}

typedef float v2f __attribute__((ext_vector_type(2)));
typedef float v8f __attribute__((ext_vector_type(8)));
typedef int v4i_t __attribute__((vector_size(4 * sizeof(int))));
typedef __attribute__((address_space(1))) v4i_t* as1_v4i_ptr;
typedef __attribute__((address_space(3))) v4i_t* as3_v4i_ptr;

#if defined(__AMDGCN__) && __has_builtin(__builtin_amdgcn_global_load_async_to_lds_b128)
#define HYENA_ASYNC 1
#endif
#if defined(__AMDGCN__) && __has_builtin(__builtin_amdgcn_wmma_f32_16x16x4_f32)
#define HYENA_WMMA 1
#endif

__device__ __forceinline__ void async_cp16(const float* gp, float* lp) {
#ifdef HYENA_ASYNC
  __builtin_amdgcn_global_load_async_to_lds_b128(
      (as1_v4i_ptr)(uintptr_t)gp,
      (as3_v4i_ptr)(uintptr_t)(uint32_t)(uintptr_t)lp,
      /*imm offset=*/0, /*cpol=*/0);
#else
  *(float4*)lp = *(const float4*)gp;
#endif
}

template <int N>
__device__ __forceinline__ void wait_async() {
#if defined(__AMDGCN__)
#if __has_builtin(__builtin_amdgcn_s_wait_asynccnt)
  __builtin_amdgcn_s_wait_asynccnt(N);
#else
  asm volatile("s_wait_asynccnt %0" ::"i"(N) : "memory");
#endif
  asm volatile("" ::: "memory");
#endif
}

struct __attribute__((aligned(16))) WaveLds {
  float zbuf[32 * 16];          // ring of z rows (l & 31), 16 cols   2 KB
  float stage[2][3][16 * 16];   // [slot][x1|x2|v][row*16+col]        6 KB
};

// conv-as-GEMM: Y[16l x 16dg] = T[16l x 24] @ Zext[24 x 16dg],
// T[i,j] = w[j-i] for 0<=j-i<=6 else 0; Zext[j] = z[l0-6+j] (rows 22,23 = 0)
__global__ __launch_bounds__(kTPB) void hyena_wmma_kernel(
    const float* __restrict__ x1, const float* __restrict__ x2,
    const float* __restrict__ v,  const float* __restrict__ cw,
    const float* __restrict__ cbias, float* __restrict__ out) {
  __shared__ WaveLds lds[kWavesPerBlock];   // 64 KB / block

  const int lane = threadIdx.x & 31;
  const int wv   = threadIdx.x >> 5;
  const int n    = lane & 15;   // A: row M; B/C/D: col N
  const int h    = lane >> 4;   // lane half selects K/M split

  const int gid  = blockIdx.x * kWavesPerBlock + wv;   // 4096 waves
  const int b    = gid >> 11;
  const int g    = (gid >> 3) & (kG - 1);
  const int seg  = gid & (kSegs - 1);
  const int lbase = seg * kSegRows;

  WaveLds& W = lds[wv];

  float w[kK];
#pragma unroll
  for (int k = 0; k < kK; ++k) w[k] = cw[g * kK + k];
  const float bias = cbias[g];

  // A-matrix (16x4 f32 layout): lane m=n, half h -> v0=A[m,2h], v1=A[m,2h+1]
  v2f A[6];
#pragma unroll
  for (int c = 0; c < 6; ++c) {
#pragma unroll
    for (int t = 0; t < 2; ++t) {
      const int idx = 4 * c + 2 * h + t - n;   // j - i into the 7-tap band
      float val = 0.0f;
#pragma unroll
      for (int k = 0; k < kK; ++k) val = (idx == k) ? w[k] : val;
      A[c][t] = val;
    }
  }

  // addressing: element (b, l, g, c) at ((b*L + l)*D + g*16 + c)
  const size_t colbase = ((size_t)b * kL) * kD + (size_t)g * kDG + (size_t)n;
  const size_t rowbase = ((size_t)b * kL) * kD + (size_t)g * kDG;

  // stage one 16-row tile of x1/x2/v: 2 async b128 per stream (512B each)
  auto stage_issue = [&](int slot, int l0) {
#pragma unroll
    for (int k = 0; k < 2; ++k) {
      const int row = k * 8 + (lane >> 2);
      const int col = (lane & 3) << 2;
      const size_t goff = rowbase + (size_t)(l0 + row) * kD + (size_t)col;
      const int loff = row * 16 + col;
      async_cp16(x1 + goff, &W.stage[slot][0][loff]);
      async_cp16(x2 + goff, &W.stage[slot][1][loff]);
      async_cp16(v  + goff, &W.stage[slot][2][loff]);
    }
  };

  // prefill causal history rows lbase-6 .. lbase-1 into the z ring
  if (h == 0) {
#pragma unroll
    for (int r = 0; r < kK - 1; ++r) {
      const int l = lbase - (kK - 1) + r;
      float z = 0.0f;
      if (l >= 0) {
        const size_t off = colbase + (size_t)l * kD;
        z = x2[off] * v[off];
      }
      W.zbuf[(l & 31) * 16 + n] = z;
    }
  }

  stage_issue(0, lbase);   // prologue

  for (int t = 0; t < kTilesPerSeg; ++t) {
    const int l0 = lbase + t * 16;
    const int s  = t & 1;
    if (t + 1 < kTilesPerSeg) {
      stage_issue(s ^ 1, l0 + 16);
      wait_async<6>();   // async loads complete in order -> slot s landed
    } else {
      wait_async<0>();
    }

    const float* stx1 = W.stage[s][0];
    const float* stx2 = W.stage[s][1];
    const float* stv  = W.stage[s][2];

    v8f acc = {0.f, 0.f, 0.f, 0.f, 0.f, 0.f, 0.f, 0.f};
#pragma unroll
    for (int c = 0; c < 6; ++c) {
      // B-matrix (4x16 f32): half h -> v0=B[2h,n], v1=B[2h+1,n]
      v2f Bv;
#pragma unroll
      for (int tt = 0; tt < 2; ++tt) {
        const int j = 4 * c + 2 * h + tt;   // Zext row index (per lane half)
        float z = 0.0f;
        if (j < kK - 1) {                   // history rows from the z ring
          const int l = l0 - (kK - 1) + j;
          z = W.zbuf[(l & 31) * 16 + n];
        } else if (j < 22) {                // fresh rows: z = x2*v from stage
          const int r = j - (kK - 1);
          z = stx2[r * 16 + n] * stv[r * 16 + n];
          W.zbuf[((l0 + r) & 31) * 16 + n] = z;
        }                                   // rows 22,23: zero pad
        Bv[tt] = z;
      }
#ifdef HYENA_WMMA
      acc = __builtin_amdgcn_wmma_f32_16x16x4_f32(
          false, A[c], false, Bv, (short)0, acc, false, false);
#else
      acc[0] += A[c][0] * Bv[0] + A[c][1] * Bv[1];  // host-pass placeholder
#endif
    }

    // epilogue: out = x1 * (Y + bias * z); D layout: vgpr r -> row r + 8h
#pragma unroll
    for (int r = 0; r < 8; ++r) {
      const int m = r + 8 * h;
      const int l = l0 + m;
      const float zv  = W.zbuf[(l & 31) * 16 + n];
      const float x1v = stx1[m * 16 + n];
      const size_t off = colbase + (size_t)l * kD;
      out[off] = x1v * (acc[r] + bias * zv);
    }
  }
}

extern "C" void kernel_launch(void* const* d_in, const int* in_sizes, int n_in,
                              void* d_out, int out_size, void* d_ws, size_t ws_size,
                              hipStream_t stream) {
  (void)in_sizes; (void)n_in; (void)out_size; (void)d_ws; (void)ws_size;
  const float* x1 = (const float*)d_in[0];
  const float* x2 = (const float*)d_in[1];
  const float* v  = (const float*)d_in[2];
  const float* cw = (const float*)d_in[3];   // [G, K]
  const float* cb = (const float*)d_in[4];   // [G]
  float* out = (float*)d_out;

  const int nWaves  = kB * kG * kSegs;              // 4096
  const int nBlocks = nWaves / kWavesPerBlock;      // 512
  dim3 grid(nBlocks);
  dim3 block(kTPB);
  hipLaunchKernelGGL(hyena_wmma_kernel, grid, block, 0, stream,
                     x1, x2, v, cw, cb, out);
}